// ChainCRF_6932077216278
// MI455X (gfx1250) — compile-verified
//
#include <hip/hip_runtime.h>
#include <hip/hip_bf16.h>

typedef __attribute__((ext_vector_type(2))) float v2f;
typedef __attribute__((ext_vector_type(8))) float v8f;

#define TT 1024
#define CC 48
#define SWIZ_SWAPX16 0x401F  // group-of-32: xor=0x10, or=0, and=0x1f

__global__ __launch_bounds__(32)
void viterbi_wmma_kernel(const float* __restrict__ x,
                         const float* __restrict__ trans,
                         float* __restrict__ out)
{
    __shared__ unsigned char bp[(TT - 1) * CC];   // 49,104 B backpointers
    __shared__ float fin[CC];

    const int lane  = threadIdx.x;          // 0..31, wave32
    const int l15   = lane & 15;
    const int half8 = (lane >> 4) << 3;     // 0 for lanes 0-15, 8 for 16-31
    const int b     = blockIdx.x;
    const float* __restrict__ xb = x + (size_t)b * TT * CC;

    // ---- Preload transition as 3x3 WMMA C-operand tiles (16x16 f32 layout):
    // VGPR r: lanes 0-15 -> (M=r, N=lane), lanes 16-31 -> (M=r+8, N=lane-16)
    v8f ctile[3][3];
#pragma unroll
    for (int ti = 0; ti < 3; ++ti)
#pragma unroll
        for (int ct = 0; ct < 3; ++ct)
#pragma unroll
            for (int r = 0; r < 8; ++r) {
                int row = ti * 16 + r + half8;
                int col = ct * 16 + l15;
                ctile[ti][ct][r] = trans[row * CC + col];
            }

    // B matrix: all entries 0.25 -> sum over K=4 gives exactly 1.0 * delta[m]
    const v2f bq = {0.25f, 0.25f};

    // delta_0 = x[0, :]; chunk ct holds delta[ct*16 + (lane&15)], replicated in both halves
    float d0 = xb[0 * CC + 0  + l15];
    float d1 = xb[0 * CC + 16 + l15];
    float d2 = xb[0 * CC + 32 + l15];

    // software-pipelined x row for t=1
    float xn0 = xb[1 * CC + 0  + l15];
    float xn1 = xb[1 * CC + 16 + l15];
    float xn2 = xb[1 * CC + 32 + l15];

    for (int t = 1; t < TT; ++t) {
        const float xc0 = xn0, xc1 = xn1, xc2 = xn2;
        if (t + 1 < TT) {
            xn0 = xb[(t + 1) * CC + 0  + l15];
            xn1 = xb[(t + 1) * CC + 16 + l15];
            xn2 = xb[(t + 1) * CC + 32 + l15];
        }
        if (t + 8 < TT)
            __builtin_prefetch(&xb[(t + 8) * CC + l15], 0, 0);  // global_prefetch

        // A operand per row-tile ti: delta[ti*16 + m] in both halves, all K slots
        const v2f a0 = {d0, d0};
        const v2f a1 = {d1, d1};
        const v2f a2 = {d2, d2};

        float nd0, nd1, nd2;
#pragma unroll
        for (int ct = 0; ct < 3; ++ct) {
            float best;
            int   bs = 0;                   // scan index 0..23 (ascending j within lane)
#pragma unroll
            for (int ti = 0; ti < 3; ++ti) {
                const v2f av = (ti == 0) ? a0 : ((ti == 1) ? a1 : a2);
                // D[m][n] = delta[ti*16+m] + trans[ti*16+m][ct*16+n]  (bit-exact f32 add)
                v8f d = __builtin_amdgcn_wmma_f32_16x16x4_f32(
                    false, av, false, bq, (short)0, ctile[ti][ct], false, false);
#pragma unroll
                for (int r = 0; r < 8; ++r) {
                    if (ti == 0 && r == 0) {        // peel: no -inf sentinel ops
                        best = d[0];
                        continue;
                    }
                    const int  s = ti * 8 + r;
                    const bool g = d[r] > best;      // strict > keeps first occurrence
                    bs   = g ? s    : bs;
                    best = g ? d[r] : best;
                }
            }
            // decode scan index -> real row j = (s&7) + (s>>3)*16 + half8
            int jmy = (bs & 7) + ((bs >> 3) << 4) + half8;

            // merge the two lane-halves (rows r vs r+8) with exact tie-break (smaller j wins)
            float ob = __int_as_float(
                __builtin_amdgcn_ds_swizzle(__float_as_int(best), SWIZ_SWAPX16));
            int   oj = __builtin_amdgcn_ds_swizzle(jmy, SWIZ_SWAPX16);
            const bool take = (ob > best) || ((ob == best) && (oj < jmy));
            best = take ? ob : best;
            jmy  = take ? oj : jmy;

            // both halves hold identical (best, jmy) for column ct*16+l15:
            // duplicate same-value LDS byte stores are benign
            bp[(t - 1) * CC + ct * 16 + l15] = (unsigned char)jmy;

            const float xr = (ct == 0) ? xc0 : ((ct == 1) ? xc1 : xc2);
            const float nv = xr + best;
            if (ct == 0) nd0 = nv; else if (ct == 1) nd1 = nv; else nd2 = nv;
        }
        d0 = nd0; d1 = nd1; d2 = nd2;
    }

    // ---- final argmax over delta_{T-1} (first occurrence) ----
    fin[0  + l15] = d0;
    fin[16 + l15] = d1;
    fin[32 + l15] = d2;
    __syncthreads();   // single-wave WG: ensures ds stores visible cross-lane

    int last = 0;
    float bv = fin[0];
#pragma unroll 4
    for (int j = 1; j < CC; ++j) {
        const float v = fin[j];
        if (v > bv) { bv = v; last = j; }
    }

    // ---- backtrack + one-hot output (lanes 0..23 each own 2 columns) ----
    float* __restrict__ ob_ = out + (size_t)b * TT * CC;
    const int c0 = 2 * lane;
    int state = last;
    if (lane < 24) {
        float2 v;
        v.x = (c0     == state) ? 1.0f : 0.0f;
        v.y = (c0 + 1 == state) ? 1.0f : 0.0f;
        *(float2*)&ob_[(size_t)(TT - 1) * CC + c0] = v;
    }
    for (int t = TT - 2; t >= 0; --t) {
        state = bp[t * CC + state];   // uniform address -> LDS broadcast
        if (lane < 24) {
            float2 v;
            v.x = (c0     == state) ? 1.0f : 0.0f;
            v.y = (c0 + 1 == state) ? 1.0f : 0.0f;
            *(float2*)&ob_[(size_t)t * CC + c0] = v;
        }
    }
}

extern "C" void kernel_launch(void* const* d_in, const int* in_sizes, int n_in,
                              void* d_out, int out_size, void* d_ws, size_t ws_size,
                              hipStream_t stream) {
    const float* x     = (const float*)d_in[0];   // [B, T, C] f32
    const float* trans = (const float*)d_in[1];   // [C, C] f32
    float* out         = (float*)d_out;           // [B, T, C] f32 one-hot

    const int B = in_sizes[0] / (TT * CC);
    viterbi_wmma_kernel<<<B, 32, 0, stream>>>(x, trans, out);
}